// QCNN_15221364097352
// MI455X (gfx1250) — compile-verified
//
#include <hip/hip_runtime.h>
#include <hip/hip_bf16.h>
#include <math.h>

// ---------------------------------------------------------------------------
// QCNN: 8-qubit state-vector circuits (13248 of them) + small MLP head.
// Circuit kernel: 1 block (256 thr) per circuit, state in LDS.
// MLP kernel: fp32 WMMA (v_wmma_f32_16x16x4_f32) per 16-row stripe.
// ---------------------------------------------------------------------------

#define NQ 8
#define DIM 256
#define NNODES 207
#define NB 64
#define NCIRC (NB * NNODES)   // 13248 = 828 * 16

typedef float v2f __attribute__((ext_vector_type(2)));
typedef float v8f __attribute__((ext_vector_type(8)));

// ---------------- quantum circuit helpers (state in LDS) -------------------

__device__ __forceinline__ void apply_ry(float2* st, int q, float theta, int t) {
    const int b = 7 - q;                 // bit position of qubit q in flat idx
    float h = 0.5f * theta;
    float c = cosf(h), s = sinf(h);
    float2 v0, v1;
    int i0 = 0, i1 = 0;
    if (t < 128) {
        int low  = t & ((1 << b) - 1);
        i0 = ((t >> b) << (b + 1)) | low;
        i1 = i0 | (1 << b);
        v0 = st[i0];
        v1 = st[i1];
    }
    __syncthreads();
    if (t < 128) {
        st[i0] = make_float2(c * v0.x - s * v1.x, c * v0.y - s * v1.y);
        st[i1] = make_float2(s * v0.x + c * v1.x, s * v0.y + c * v1.y);
    }
    __syncthreads();
}

__device__ __forceinline__ void apply_rz(float2* st, int q, float theta, int t) {
    // diagonal: bit==0 -> *(c - i s), bit==1 -> *(c + i s), h = theta/2
    const int b = 7 - q;
    float h = 0.5f * theta;
    float c = cosf(h), s = sinf(h);
    float2 a = st[t];
    float sg = ((t >> b) & 1) ? s : -s;
    st[t] = make_float2(c * a.x - sg * a.y, c * a.y + sg * a.x);
    __syncthreads();
}

__device__ __forceinline__ void apply_cnot(float2* st, int cq, int tq, int t) {
    const int bc = 7 - cq, bt = 7 - tq;
    int src = ((t >> bc) & 1) ? (t ^ (1 << bt)) : t;
    float2 v = st[src];
    __syncthreads();
    st[t] = v;
    __syncthreads();
}

__device__ __forceinline__ void conv_pair(float2* st, int q1, int q2,
                                          const float* p, int t) {
    apply_ry(st, q1, p[0], t);
    apply_ry(st, q2, p[1], t);
    apply_cnot(st, q1, q2, t);
    apply_ry(st, q1, p[2], t);
    apply_ry(st, q2, p[3], t);
    apply_cnot(st, q2, q1, t);
}

// ---------------- kernel 1: one circuit per block --------------------------

__global__ __launch_bounds__(256)
void qcnn_circuit_kernel(const float* __restrict__ x,      // (64,12,207,2)
                         const float* __restrict__ adj,    // (207,207)
                         const float* __restrict__ w_proj, // (8,24)
                         const float* __restrict__ b_proj, // (8)
                         const float* __restrict__ qp,     // (64)
                         float* __restrict__ z_out)        // (13248,8)
{
    __shared__ float2 st[DIM];
    __shared__ float  ang[NQ];
    __shared__ float  qps[64];
    __shared__ float  zacc[NQ];

    const int g  = blockIdx.x;        // circuit index = b*207 + nn
    const int bb = g / NNODES;
    const int nn = g % NNODES;
    const int t  = threadIdx.x;       // 0..255

    if (t < 64) qps[t] = qp[t];
    if (t < NQ) {
        // angles[q] = b_proj[q] + sum_{s,f} x[b,s,nn,f] * w_proj[q, s*2+f]
        float acc = b_proj[t];
        #pragma unroll
        for (int ss = 0; ss < 12; ++ss) {
            #pragma unroll
            for (int f = 0; f < 2; ++f) {
                float xv = x[(((size_t)bb * 12 + ss) * NNODES + nn) * 2 + f];
                acc = fmaf(xv, w_proj[t * 24 + ss * 2 + f], acc);
            }
        }
        const float PI = 3.14159265358979323846f;
        ang[t]  = fminf(fmaxf(acc, -PI), PI);
        zacc[t] = 0.0f;
    }
    st[t] = make_float2((t == 0) ? 1.0f : 0.0f, 0.0f);
    __syncthreads();

    // initial rotations
    for (int i = 0; i < NQ; ++i)      apply_ry(st, i, ang[i], t);
    for (int i = 0; i < NQ / 2; ++i)  apply_rz(st, i, ang[i + NQ / 2], t);

    // adjacency-conditioned permutation (conditions are block-uniform)
    const int c = nn % NQ;
    for (int tt = 0; tt < NQ; ++tt) {
        bool cond = (adj[(size_t)c * NNODES + tt] > 0.0f) && (tt != c);
        if (cond) {
            int bitc = (t >> (7 - c)) & 1;
            int src  = t ^ (bitc << (7 - tt));
            float2 v = st[src];
            __syncthreads();
            st[t] = v;
            __syncthreads();
        }
    }

    // conv layers
    int off = 0;
    for (int layer = 0; layer < 2; ++layer) {
        int k = off;
        for (int i = 0; i < NQ - 1; i += 2) { conv_pair(st, i, i + 1, qps + k, t); k += 4; }
        for (int i = 1; i < NQ - 1; i += 2) { conv_pair(st, i, i + 1, qps + k, t); k += 4; }
        off += 4 * (NQ - 1);
    }

    // pooling rotations on even qubits
    for (int q = 0; q < NQ; q += 2) apply_ry(st, q, qps[off + q], t);

    // Z expectations: z[w] = sum_{bit_w==0} |amp|^2 - sum_{bit_w==1} |amp|^2
    float2 a = st[t];
    float p = a.x * a.x + a.y * a.y;
    #pragma unroll
    for (int w = 0; w < NQ; ++w) {
        float sgn = ((t >> (7 - w)) & 1) ? -p : p;
        atomicAdd(&zacc[w], sgn);     // ds_add_f32
    }
    __syncthreads();
    if (t < NQ) z_out[(size_t)g * NQ + t] = zacc[t];
}

// ---------------- kernel 2: MLP head with fp32 WMMA ------------------------
// z (13248,8) -> relu(z@w1.T+b1) (.,64) -> relu(@w2.T+b2) (.,32) -> @w3.T+b3
// One wave32 per 16-row stripe. 828 blocks.

__global__ __launch_bounds__(32)
void qcnn_mlp_kernel(const float* __restrict__ z,   // (13248,8)
                     const float* __restrict__ w1,  // (64,8)
                     const float* __restrict__ b1,  // (64)
                     const float* __restrict__ w2,  // (32,64)
                     const float* __restrict__ b2,  // (32)
                     const float* __restrict__ w3,  // (1,32)
                     const float* __restrict__ b3,  // (1)
                     float* __restrict__ out)       // (13248)
{
    __shared__ float h1[16][65];   // padded vs bank conflicts
    __shared__ float h2[16][33];

    const int lane  = threadIdx.x;        // 0..31
    const int m     = lane & 15;          // row (A) / col (B,C,D) within tile
    const int kbase = (lane >> 4) * 2;    // k-pair select for A/B fragments
    const int hi    = lane >> 4;          // C/D row offset select
    const int row0  = blockIdx.x * 16;

    // ---- layer 1: (16x8) @ (8x64) via 2 k-steps x 4 N-tiles ----
    v8f acc[4];
    #pragma unroll
    for (int nt = 0; nt < 4; ++nt) acc[nt] = {};

    #pragma unroll
    for (int kc = 0; kc < 2; ++kc) {
        v2f afr;
        afr.x = z[(size_t)(row0 + m) * 8 + kc * 4 + kbase + 0];
        afr.y = z[(size_t)(row0 + m) * 8 + kc * 4 + kbase + 1];
        #pragma unroll
        for (int nt = 0; nt < 4; ++nt) {
            int n = nt * 16 + m;          // B column striped across lanes
            v2f bfr;
            bfr.x = w1[n * 8 + kc * 4 + kbase + 0];
            bfr.y = w1[n * 8 + kc * 4 + kbase + 1];
            acc[nt] = __builtin_amdgcn_wmma_f32_16x16x4_f32(
                false, afr, false, bfr, (short)0, acc[nt], false, false);
        }
    }
    // bias + relu, stage to LDS in (row, col) layout
    #pragma unroll
    for (int nt = 0; nt < 4; ++nt) {
        int n = nt * 16 + m;
        float bias = b1[n];
        #pragma unroll
        for (int r = 0; r < 8; ++r)
            h1[r + 8 * hi][n] = fmaxf(acc[nt][r] + bias, 0.0f);
    }
    __syncthreads();

    // ---- layer 2: (16x64) @ (64x32) via 16 k-steps x 2 N-tiles ----
    v8f acc2[2];
    acc2[0] = {};
    acc2[1] = {};
    #pragma unroll
    for (int kc = 0; kc < 16; ++kc) {
        v2f afr;
        afr.x = h1[m][kc * 4 + kbase + 0];
        afr.y = h1[m][kc * 4 + kbase + 1];
        #pragma unroll
        for (int nt = 0; nt < 2; ++nt) {
            int n = nt * 16 + m;
            v2f bfr;
            bfr.x = w2[n * 64 + kc * 4 + kbase + 0];
            bfr.y = w2[n * 64 + kc * 4 + kbase + 1];
            acc2[nt] = __builtin_amdgcn_wmma_f32_16x16x4_f32(
                false, afr, false, bfr, (short)0, acc2[nt], false, false);
        }
    }
    #pragma unroll
    for (int nt = 0; nt < 2; ++nt) {
        int n = nt * 16 + m;
        float bias = b2[n];
        #pragma unroll
        for (int r = 0; r < 8; ++r)
            h2[r + 8 * hi][n] = fmaxf(acc2[nt][r] + bias, 0.0f);
    }
    __syncthreads();

    // ---- layer 3: (16x32) @ (32x1), scalar dot per row ----
    if (lane < 16) {
        float s = b3[0];
        #pragma unroll
        for (int n = 0; n < 32; ++n)
            s = fmaf(h2[lane][n], w3[n], s);
        out[row0 + lane] = s;
    }
}

// ---------------------------------------------------------------------------

extern "C" void kernel_launch(void* const* d_in, const int* in_sizes, int n_in,
                              void* d_out, int out_size, void* d_ws, size_t ws_size,
                              hipStream_t stream) {
    const float* x      = (const float*)d_in[0];
    const float* adj    = (const float*)d_in[1];
    const float* w_proj = (const float*)d_in[2];
    const float* b_proj = (const float*)d_in[3];
    const float* qp     = (const float*)d_in[4];
    const float* w1     = (const float*)d_in[5];
    const float* b1     = (const float*)d_in[6];
    const float* w2     = (const float*)d_in[7];
    const float* b2     = (const float*)d_in[8];
    const float* w3     = (const float*)d_in[9];
    const float* b3     = (const float*)d_in[10];
    float* out = (float*)d_out;
    float* z   = (float*)d_ws;        // (13248, 8) scratch = 424 KB

    qcnn_circuit_kernel<<<NCIRC, 256, 0, stream>>>(x, adj, w_proj, b_proj, qp, z);
    qcnn_mlp_kernel<<<NCIRC / 16, 32, 0, stream>>>(z, w1, b1, w2, b2, w3, b3, out);
}